// SageBlock_45578192945252
// MI455X (gfx1250) — compile-verified
//
#include <hip/hip_runtime.h>
#include <math.h>

// ---------------------------------------------------------------------------
// SAGEConv(mean, no root, no bias) + ELU + BatchNorm1d (training stats)
//
// MI455X reasoning:
//  * Edge aggregation dominates: 1.2M gathers + 76.8M f32 atomics; x (25.6MB)
//    and agg (25.6MB) are L2-resident (192MB L2) -> atomics/gathers at L2 BW.
//  * GEMM (0.82 GFLOP) goes through fp32 WMMA (v_wmma_f32_16x16x4_f32).
//    Each wave owns ONE 16-col tile so its 16 loop-invariant B frags (32
//    VGPRs) stay register-resident; inner loop is global_load_b64(A) ->
//    v_pk_mul (mean) -> v_wmma, fully pipelined, no spills.
//  * Store epilogue uses one per-lane base pointer + immediate row offsets
//    (256B stride fits the 24-bit IOFFSET) instead of per-row 64-bit math.
//  * BN stats fused in the GEMM epilogue, finalized + applied in-place.
// ---------------------------------------------------------------------------

#define N_NODES 100000
#define N_EDGES 1200000
#define D       64
#define N_TILES (N_NODES / 16)   // 6250, exact
#define BN_EPS  1e-5f

typedef __attribute__((ext_vector_type(2))) float v2f;
typedef __attribute__((ext_vector_type(8))) float v8f;

// ---------------------------------------------------------------------------
// Kernel 1: scatter-add aggregation.  64 threads per edge (one per feature).
// ---------------------------------------------------------------------------
__global__ void sage_agg_kernel(const float* __restrict__ x,
                                const int*   __restrict__ ei,
                                float* __restrict__ agg,
                                float* __restrict__ cnt) {
    unsigned gid = blockIdx.x * 256u + threadIdx.x;
    unsigned e = gid >> 6;        // edge id
    unsigned f = gid & 63u;       // feature id
    if (e >= N_EDGES) return;
    int src = ei[e];              // edge_index row 0
    int dst = ei[N_EDGES + e];    // edge_index row 1
    float v = x[(size_t)src * D + f];
    atomicAdd(&agg[(size_t)dst * D + f], v);
    if (f == 0) atomicAdd(&cnt[dst], 1.0f);
}

// ---------------------------------------------------------------------------
// Kernel 2: mean-divide + GEMM (fp32 WMMA 16x16x4) + ELU + BN partial stats.
// Each wave owns one 16-wide column tile (c in 0..3) and grid-strides over
// the 6250 row tiles.  B fragments (16 x v2f) are loaded once per wave and
// stay in registers; accumulator is a single v8f.
// ---------------------------------------------------------------------------
__global__ void sage_gemm_bn_kernel(const float* __restrict__ agg,
                                    const float* __restrict__ cnt,
                                    const float* __restrict__ W,
                                    float* __restrict__ h,        // = d_out
                                    float* __restrict__ sums,
                                    float* __restrict__ sumsq,
                                    int totalWaves) {
    const int lane = threadIdx.x & 31;
    const int wave = blockIdx.x * (blockDim.x >> 5) + (threadIdx.x >> 5);
    const int lm = lane & 15;   // A-row / B-col / C-col within tile
    const int kh = lane >> 4;   // K-half selector (A/B), M-half selector (C/D)

    const int c        = wave & 3;          // column tile 0..3
    const int tile0    = wave >> 2;         // first row tile
    const int tileStep = totalWaves >> 2;   // row-tile stride
    const int n        = c * 16 + lm;       // this lane's feature column

    // Load the 16 loop-invariant B fragments for column tile c.
    // breg[j] covers K rows [4j + 2kh, 4j + 2kh + 1].
    v2f breg[16];
    #pragma unroll
    for (int j = 0; j < 16; ++j) {
        const int krow = 4 * j + 2 * kh;
        breg[j].x = W[krow * D + n];
        breg[j].y = W[(krow + 1) * D + n];
    }

    float sSum = 0.f, sSq = 0.f;
    const v8f acc0 = {0.f, 0.f, 0.f, 0.f, 0.f, 0.f, 0.f, 0.f};

    for (int tile = tile0; tile < N_TILES; tile += tileStep) {
        const unsigned row0 = (unsigned)tile * 16u;
        const unsigned arow = row0 + (unsigned)lm;
        const float inv = 1.0f / fmaxf(cnt[arow], 1.0f);
        const float2* ap = (const float2*)(agg + (size_t)arow * D);

        v8f acc = acc0;
        #pragma unroll
        for (int j = 0; j < 16; ++j) {
            // A frag: lane holds mean_agg[row0+lm][4j + 2kh + {0,1}]
            float2 av = ap[2 * j + kh];
            v2f a;
            a.x = av.x * inv;
            a.y = av.y * inv;
            acc = __builtin_amdgcn_wmma_f32_16x16x4_f32(
                false, a, false, breg[j], (short)0, acc, false, false);
        }

        // Epilogue: ELU, store h, accumulate per-feature BN partials.
        // One base pointer per lane; row stride D*4 = 256B folds into the
        // store's immediate offset.
        float* hp = h + (size_t)(row0 + (unsigned)(kh * 8)) * D + n;
        #pragma unroll
        for (int i = 0; i < 8; ++i) {
            float v = acc[i];
            v = (v > 0.f) ? v : (__expf(v) - 1.0f);
            hp[i * D] = v;                      // global_store_b32, imm offset
            sSum += v;
            sSq  += v * v;
        }
    }

    // Fold lane pairs (L, L^16), which share the same feature column, then
    // one atomic per (wave, feature).
    float s = sSum + __shfl_xor(sSum, 16, 32);
    float q = sSq  + __shfl_xor(sSq,  16, 32);
    if (lane < 16) {
        atomicAdd(&sums[n],  s);
        atomicAdd(&sumsq[n], q);
    }
}

// ---------------------------------------------------------------------------
// Kernel 3: finalize BN stats -> scale/shift (64 threads).
// ---------------------------------------------------------------------------
__global__ void sage_bn_stats_kernel(const float* __restrict__ sums,
                                     const float* __restrict__ sumsq,
                                     const float* __restrict__ gamma,
                                     const float* __restrict__ beta,
                                     float* __restrict__ scale,
                                     float* __restrict__ shift) {
    int f = threadIdx.x;
    if (f < D) {
        const float invN = 1.0f / (float)N_NODES;
        float mu  = sums[f] * invN;
        float var = sumsq[f] * invN - mu * mu;     // biased, matches jnp.var
        float rs  = rsqrtf(var + BN_EPS);
        float sc  = rs * gamma[f];
        scale[f] = sc;
        shift[f] = beta[f] - mu * sc;
    }
}

// ---------------------------------------------------------------------------
// Kernel 4: apply BN in place on d_out, float4-vectorized.
// ---------------------------------------------------------------------------
__global__ void sage_bn_apply_kernel(float* __restrict__ out,
                                     const float* __restrict__ scale,
                                     const float* __restrict__ shift) {
    unsigned idx = blockIdx.x * 256u + threadIdx.x;   // float4 index
    if (idx >= (N_NODES * D) / 4) return;
    const float4* sc4 = (const float4*)scale;
    const float4* sh4 = (const float4*)shift;
    float4* o4 = (float4*)out;
    const int fg = idx & 15;                          // D/4 = 16 groups
    float4 v = o4[idx];
    float4 sc = sc4[fg];
    float4 sh = sh4[fg];
    v.x = v.x * sc.x + sh.x;
    v.y = v.y * sc.y + sh.y;
    v.z = v.z * sc.z + sh.z;
    v.w = v.w * sc.w + sh.w;
    o4[idx] = v;
}

// ---------------------------------------------------------------------------
extern "C" void kernel_launch(void* const* d_in, const int* in_sizes, int n_in,
                              void* d_out, int out_size, void* d_ws, size_t ws_size,
                              hipStream_t stream) {
    const float* x     = (const float*)d_in[0];   // [100000,64]
    const int*   ei    = (const int*)  d_in[1];   // [2,1200000]
    const float* W     = (const float*)d_in[2];   // [64,64]
    const float* gamma = (const float*)d_in[3];   // [64]
    const float* beta  = (const float*)d_in[4];   // [64]
    float* out = (float*)d_out;                   // [100000,64], also holds h
    float* ws  = (float*)d_ws;

    // Workspace layout (floats):
    float* agg   = ws;                       // 6,400,000
    float* cnt   = ws + (size_t)N_NODES * D; //   100,000
    float* sums  = cnt + N_NODES;            //        64
    float* sumsq = sums + D;                 //        64
    float* scale = sumsq + D;                //        64
    float* shift = scale + D;                //        64

    // Zero the accumulators every call (graph-replay safe).
    size_t zero_bytes = ((size_t)N_NODES * D + N_NODES + 2 * D) * sizeof(float);
    hipMemsetAsync(agg, 0, zero_bytes, stream);

    // 1) segment-sum aggregation + degree counts
    {
        unsigned total = (unsigned)N_EDGES * 64u;
        sage_agg_kernel<<<(total + 255u) / 256u, 256, 0, stream>>>(x, ei, agg, cnt);
    }

    // 2) mean + GEMM (fp32 WMMA) + ELU + BN partial sums
    {
        const int blocks = 256;                 // 2048 waves: 512 per col tile
        const int waves  = blocks * (256 / 32);
        sage_gemm_bn_kernel<<<blocks, 256, 0, stream>>>(agg, cnt, W, out,
                                                        sums, sumsq, waves);
    }

    // 3) finalize stats
    sage_bn_stats_kernel<<<1, 64, 0, stream>>>(sums, sumsq, gamma, beta,
                                               scale, shift);

    // 4) apply BN in place
    {
        const unsigned nf4 = (N_NODES * D) / 4; // 1,600,000
        sage_bn_apply_kernel<<<(nf4 + 255u) / 256u, 256, 0, stream>>>(out, scale,
                                                                      shift);
    }
}